// HMM_82291573391586
// MI455X (gfx1250) — compile-verified
//
#include <hip/hip_runtime.h>
#include <hip/hip_bf16.h>

typedef __attribute__((ext_vector_type(2))) float v2f;
typedef __attribute__((ext_vector_type(8))) float v8f;

#define BIG_NEG   (-1e10f)
#define LN2f      0.6931471805599453f
#define LOG_2PIf  1.8378770664093453f

// per-element Gaussian log-density term (without the -0.5*log(2pi) constant)
__device__ __forceinline__ float lp_term(float x, float mu, float sd) {
    float inv = __builtin_amdgcn_rcpf(sd);          // v_rcp_f32
    float z   = (x - mu) * inv;
    // -0.5*z^2 - ln(sd);  v_log_f32 is log2 -> scale by ln2
    return fmaf(-0.5f * z, z, -LN2f * __builtin_amdgcn_logf(sd));
}

// Emission kernel: one block per (t,b). 256 threads = 8 waves; wave w owns
// states n in [16w,16w+16). WMMA A-matrix 16x4 f32 layout (ISA 7.12.2):
//   lanes 0-15  : row M=lane,    VGPR0=K0, VGPR1=K1
//   lanes 16-31 : row M=lane-16, VGPR0=K2, VGPR1=K3
// Each K-partial covers M/4=20 features. D = A x ones(4x16) puts row-sums in
// accumulator VGPRs 0..7 of lane 0 (rows 0-7) and lane 16 (rows 8-15).
__global__ __launch_bounds__(256) void hmm_emissions(
    const float* __restrict__ mel,    // (B,M,T)
    const float* __restrict__ means,  // (B,T,N,M)
    const float* __restrict__ stds,   // (B,T,N,M)
    const int*   __restrict__ tlens,  // (B,)
    float*       __restrict__ emis,   // (B,T,N)
    int Bn, int Mdim, int Tn, int Nn)
{
    const int t    = blockIdx.x;
    const int b    = blockIdx.y;
    const int tid  = threadIdx.x;
    const int wave = tid >> 5;
    const int lane = tid & 31;
    const int n    = (wave << 4) + (lane & 15);
    const int chunk = Mdim >> 2;                   // 20 features per K-partial
    const int kbase = (lane < 16) ? 0 : 2;

    __shared__ float x_s[128];
    for (int i = tid; i < Mdim; i += blockDim.x)
        x_s[i] = mel[(size_t)(b * Mdim + i) * Tn + t];
    __syncthreads();

    const size_t rowbase = ((size_t)(b * Tn + t) * Nn + n) * Mdim;

    v2f a;
    #pragma unroll
    for (int kk = 0; kk < 2; ++kk) {
        const int m0 = (kbase + kk) * chunk;       // 16B-aligned (20*4=80B)
        const float4* mu4 = (const float4*)(means + rowbase + m0);
        const float4* sd4 = (const float4*)(stds  + rowbase + m0);
        float acc = 0.0f;
        #pragma unroll
        for (int j = 0; j < 5; ++j) {              // 5 x float4 = 20 floats
            float4 mu = mu4[j];
            float4 sd = sd4[j];
            const float* xs = &x_s[m0 + 4 * j];
            acc += lp_term(xs[0], mu.x, sd.x);
            acc += lp_term(xs[1], mu.y, sd.y);
            acc += lp_term(xs[2], mu.z, sd.z);
            acc += lp_term(xs[3], mu.w, sd.w);
        }
        a[kk] = acc;
    }

    // Row-sum the 4 K-partials of each of the wave's 16 states via WMMA.
    v2f bones; bones.x = 1.0f; bones.y = 1.0f;
    v8f c = {};
    v8f d = __builtin_amdgcn_wmma_f32_16x16x4_f32(
        /*neg_a=*/false, a, /*neg_b=*/false, bones,
        /*c_mod=*/(short)0, c, /*reuse_a=*/false, /*reuse_b=*/false);

    const int tlen = tlens[b];
    if ((lane & 15) == 0) {
        const int rbase = (wave << 4) + ((lane < 16) ? 0 : 8);
        const float cst = -0.5f * LOG_2PIf * (float)Mdim;
        #pragma unroll
        for (int g = 0; g < 8; ++g) {
            const int nn = rbase + g;
            float e = d[g] + cst;
            emis[(size_t)(b * Tn + t) * Nn + nn] = (nn < tlen) ? e : 0.0f;
        }
    }
}

// Sequential HMM forward scan: one block per batch element, N threads.
__global__ __launch_bounds__(128) void hmm_scan(
    const float* __restrict__ emis,   // (B,T,N)
    const float* __restrict__ tv,     // (B,T,N)
    const int*   __restrict__ tlens,  // (B,)
    const int*   __restrict__ mlens,  // (B,)
    float*       __restrict__ out,    // (B,)
    int Tn, int Nn)
{
    const int b = blockIdx.x;
    const int n = threadIdx.x;

    __shared__ float s_alpha[256];
    __shared__ float s_v[256];
    __shared__ float s_red[256];

    const int  tlen  = tlens[b];
    const int  mlen  = mlens[b];
    const bool smask = n < tlen;

    float sum_log_c  = 0.0f;
    float last_alpha = BIG_NEG;
    float last_w     = 0.0f;

    s_alpha[n] = 0.0f;
    __syncthreads();

    for (int t = 0; t < Tn; ++t) {
        const size_t idx = (size_t)(b * Tn + t) * Nn + n;
        const float em = emis[idx];
        const float w  = tv[idx];
        const float tp = 1.0f / (1.0f + expf(-w));
        const float ls = logf(fmaxf(1.0f - tp, 1e-8f));  // log(1-p) clamped
        const float lt = logf(fmaxf(tp, 1e-8f));         // log(p)   clamped

        const float ap = s_alpha[n];
        s_v[n] = ap + lt;
        __syncthreads();

        const float staying = ap + ls;
        const float leaving = (n == 0) ? BIG_NEG : s_v[n - 1];
        const float mx = fmaxf(staying, leaving);
        float ov = mx + logf(expf(staying - mx) + expf(leaving - mx));
        ov = smask ? ov : BIG_NEG;

        const float temp = (t == 0) ? (((n == 0) ? 0.0f : BIG_NEG) + em)
                                    : (em + ov);

        // logsumexp over N: max pass
        s_red[n] = temp;
        __syncthreads();
        for (int s = Nn >> 1; s > 0; s >>= 1) {
            if (n < s) s_red[n] = fmaxf(s_red[n], s_red[n + s]);
            __syncthreads();
        }
        const float m = s_red[0];
        __syncthreads();
        // sum pass
        s_red[n] = expf(temp - m);
        __syncthreads();
        for (int s = Nn >> 1; s > 0; s >>= 1) {
            if (n < s) s_red[n] += s_red[n + s];
            __syncthreads();
        }
        const float log_c = m + logf(s_red[0]);
        __syncthreads();

        const float al = temp - log_c;
        s_alpha[n] = al;
        if (t < mlen)      sum_log_c += log_c;
        if (t == mlen - 1) { last_alpha = al; last_w = w; }
        __syncthreads();
    }

    // absorption term
    const float la  = smask ? last_alpha : BIG_NEG;
    const float tpl = 1.0f / (1.0f + expf(-last_w));
    const float lpv = (n == tlen - 1) ? logf(fmaxf(tpl, 1e-8f)) : BIG_NEG;
    const float fin = fmaxf(la + lpv, 1e-8f);

    s_red[n] = fin;
    __syncthreads();
    for (int s = Nn >> 1; s > 0; s >>= 1) {
        if (n < s) s_red[n] = fmaxf(s_red[n], s_red[n + s]);
        __syncthreads();
    }
    const float m2 = s_red[0];
    __syncthreads();
    s_red[n] = expf(fin - m2);
    __syncthreads();
    for (int s = Nn >> 1; s > 0; s >>= 1) {
        if (n < s) s_red[n] += s_red[n + s];
        __syncthreads();
    }
    if (n == 0) out[b] = sum_log_c + m2 + logf(s_red[0]);
}

extern "C" void kernel_launch(void* const* d_in, const int* in_sizes, int n_in,
                              void* d_out, int out_size, void* d_ws, size_t ws_size,
                              hipStream_t stream) {
    const float* mel   = (const float*)d_in[0];  // (B,M,T)
    const float* means = (const float*)d_in[1];  // (B,T,N,M)
    const float* stds  = (const float*)d_in[2];  // (B,T,N,M)
    const float* tv    = (const float*)d_in[3];  // (B,T,N)
    const int*   tlens = (const int*)d_in[4];    // (B,)
    const int*   mlens = (const int*)d_in[5];    // (B,)
    float*       out   = (float*)d_out;          // (B,)

    const int B = in_sizes[4];
    const long long TN  = (long long)in_sizes[3] / B;
    const long long MT  = (long long)in_sizes[0] / B;
    const long long TNM = (long long)in_sizes[1] / B;
    const int M = (int)(TNM / TN);   // 80
    const int T = (int)(MT / M);     // 512
    const int N = (int)(TN / T);     // 128

    float* emis = (float*)d_ws;      // B*T*N floats = 2 MB

    dim3 gridA(T, B);
    hmm_emissions<<<gridA, dim3(N * 2), 0, stream>>>(mel, means, stds, tlens,
                                                     emis, B, M, T, N);
    hmm_scan<<<dim3(B), dim3(N), 0, stream>>>(emis, tv, tlens, mlens, out, T, N);
}